// MinimalGDAModel_14010183320207
// MI455X (gfx1250) — compile-verified
//
#include <hip/hip_runtime.h>
#include <math.h>

#define B_    32
#define S_    2048
#define IN_   300
#define NA_   64
#define GRAV  0.01f

#define ROWS_PER_BLOCK 128
#define NBLK ((B_ * S_) / ROWS_PER_BLOCK)   // 512 blocks

// d_out flat offsets (floats): af[512] | rec[32*2048*300] | masses[32] | change[1]
#define AF_OFF  0
#define REC_OFF 512
#define M_OFF   19661312
#define C_OFF   19661344

// d_ws flat offsets (floats)
#define WS_AFPART  0        // NBLK*16 = 8192
#define WS_CHGPART 8192     // NBLK    = 512
#define WS_REC     8704     // 32*300  = 9600

typedef float v2f __attribute__((ext_vector_type(2)));
typedef float v8f __attribute__((ext_vector_type(8)));

// ---- LDS layout (float offsets). Encode-phase and gravity-phase regions alias. ----
#define L_ATT   0        // 1024 : attractors [64][16]                (persistent)
#define L_AA    1024     // 64   : |attractor|^2                      (persistent)
#define L_B1S   1088     // 16   : enc bias1                          (persistent)
#define L_B2S   1104     // 16   : enc bias2                          (persistent)
#define L_W2S   1120     // 256  : enc_w2                             (persistent)
#define L_CHG   1376     // 128  : per-thread change partials         (persistent)
// ---- phase union starts at 1504 ----
#define L_W1T   1504     // 4800 : enc_w1 transposed [16][300]        (encode phase)
#define L_HS    6304     // 2176 : h scratch [4][32][17]              (encode phase)
#define L_P     1504     // 2048 : positions [4 waves][32][16]        (gravity phase)
#define L_PP    3552     // 128  : |p|^2 per sample                   (gravity phase)
#define L_S0    3680     // 128  : coef row-sums per sample           (gravity phase)
#define L_COEF  3808     // 8448 : coef [4 waves][2*16 rows][66]      (gravity phase)
#define L_TOTAL 12256    // 49,024 bytes

#define WMMA4(a, b, c) __builtin_amdgcn_wmma_f32_16x16x4_f32(false, (a), false, (b), (short)0, (c), false, false)

__global__ __launch_bounds__(128) void gda_main(
    const float* __restrict__ X,  const float* __restrict__ W1,
    const float* __restrict__ b1, const float* __restrict__ W2,
    const float* __restrict__ b2, const float* __restrict__ att,
    float* __restrict__ af_part,  float* __restrict__ chg_part)
{
  __shared__ float smem[L_TOTAL];
  const int tid = threadIdx.x;
  const int bk  = blockIdx.x;

  // ================= cooperative preload =================
  for (int i = tid; i < IN_ * 16; i += 128) {
    int kk = i >> 4, n = i & 15;
    smem[L_W1T + n * IN_ + kk] = W1[i];          // W1 is [300][16] row-major -> [16][300]
  }
  if (tid < 16) { smem[L_B1S + tid] = b1[tid]; smem[L_B2S + tid] = b2[tid]; }
  for (int i = tid; i < 256; i += 128)      smem[L_W2S + i] = W2[i];
  for (int i = tid; i < NA_ * 16; i += 128) smem[L_ATT + i] = att[i];
  __syncthreads();
  if (tid < NA_) {                               // waves 0..1 fully active
    float s = 0.f;
    #pragma unroll
    for (int d = 0; d < 16; ++d) { float v = smem[L_ATT + tid * 16 + d]; s = fmaf(v, v, s); }
    smem[L_AA + tid] = s;
  }

  const int wv = tid >> 5, ln = tid & 31;
  const int nIdx = ln & 15, hi = ln >> 4;
  const int rowBase = bk * ROWS_PER_BLOCK + wv * 32;

  // ================= encoder layer 1: WMMA f32 16x16x4, two tiles/wave =================
  const float* xr0 = X + (size_t)(rowBase + nIdx) * IN_;
  const float* xr1 = xr0 + (size_t)16 * IN_;
  const float* w1row = &smem[L_W1T + nIdx * IN_];
  v8f c0 = {}, c1 = {};
  for (int k = 0; k < IN_; k += 4) {
    int kk = k + 2 * hi;
    v2f bfr = *(const v2f*)(w1row + kk);
    v2f a0  = *(const v2f*)(xr0 + kk);
    v2f a1  = *(const v2f*)(xr1 + kk);
    c0 = WMMA4(a0, bfr, c0);
    c1 = WMMA4(a1, bfr, c1);
  }
  {
    float bn = smem[L_B1S + nIdx];
    #pragma unroll
    for (int i = 0; i < 8; ++i) {
      int m = i + 8 * hi;
      smem[L_HS + (wv * 32 + m) * 17 + nIdx]      = fmaxf(c0[i] + bn, 0.f);
      smem[L_HS + (wv * 32 + m + 16) * 17 + nIdx] = fmaxf(c1[i] + bn, 0.f);
    }
  }
  // same-wave readback (DS in-order within a wave)
  float hv[16];
  #pragma unroll
  for (int c = 0; c < 16; ++c) hv[c] = smem[L_HS + (wv * 32 + ln) * 17 + c];

  // ================= encoder layer 2 (VALU, W2 broadcast reads) =================
  float pos[16];
  #pragma unroll
  for (int d = 0; d < 16; ++d) {
    float acc = smem[L_B2S + d];
    #pragma unroll
    for (int c = 0; c < 16; ++c) acc = fmaf(hv[c], smem[L_W2S + c * 16 + d], acc);
    pos[d] = acc;
  }

  __syncthreads();   // all waves done with W1T/HS; aa visible; safe to alias -> P/COEF

  // initial positions: thread tid <-> sample tid (sample-major, stride 16, b128 stores)
  {
    float4* Pr = (float4*)&smem[L_P + tid * 16];
    float4 t0; t0.x = pos[0];  t0.y = pos[1];  t0.z = pos[2];  t0.w = pos[3];
    float4 t1; t1.x = pos[4];  t1.y = pos[5];  t1.z = pos[6];  t1.w = pos[7];
    float4 t2; t2.x = pos[8];  t2.y = pos[9];  t2.z = pos[10]; t2.w = pos[11];
    float4 t3; t3.x = pos[12]; t3.y = pos[13]; t3.z = pos[14]; t3.w = pos[15];
    Pr[0] = t0; Pr[1] = t1; Pr[2] = t2; Pr[3] = t3;
  }

  // ---- hoist loop-invariant B fragments (attractors) into registers ----
  // GEMM1: Dot[s,a] = sum_d P[s][d]*att[a][d]; B1[t][c] holds B[K=dim][N=att tile t]
  // GEMM2: Sd[s,d]  = sum_a coef[s][a]*att[a][d]; B2[c] holds B[K=att][N=dim]
  const float* atp = &smem[L_ATT];
  v2f B1[4][4], B2[16];
  #pragma unroll
  for (int t = 0; t < 4; ++t)
    #pragma unroll
    for (int c = 0; c < 4; ++c)
      B1[t][c] = *(const v2f*)(atp + (t * 16 + nIdx) * 16 + c * 4 + 2 * hi);
  #pragma unroll
  for (int c = 0; c < 16; ++c) {
    int kk = c * 4 + 2 * hi;
    v2f b; b.x = atp[kk * 16 + nIdx]; b.y = atp[(kk + 1) * 16 + nIdx];
    B2[c] = b;
  }
  float aaL[4];
  #pragma unroll
  for (int t = 0; t < 4; ++t) aaL[t] = smem[L_AA + t * 16 + nIdx];

  const int pBase = L_P + wv * 32 * 16;      // this wave's 32 position rows
  const int cBase = L_COEF + wv * 2112;      // this wave's coef rows, stride 66

  // ================= gravity: WMMA dot-products, barrier-free (per-wave state) =================
  float chg = 0.f;
  for (int it = 0; it < 20; ++it) {
    // |p|^2 for my sample (lane ln <-> sample wv*32+ln)
    {
      const float4* pr = (const float4*)&smem[pBase + ln * 16];
      float4 q0 = pr[0], q1 = pr[1], q2 = pr[2], q3 = pr[3];
      float pp = 0.f;
      pp = fmaf(q0.x, q0.x, pp); pp = fmaf(q0.y, q0.y, pp); pp = fmaf(q0.z, q0.z, pp); pp = fmaf(q0.w, q0.w, pp);
      pp = fmaf(q1.x, q1.x, pp); pp = fmaf(q1.y, q1.y, pp); pp = fmaf(q1.z, q1.z, pp); pp = fmaf(q1.w, q1.w, pp);
      pp = fmaf(q2.x, q2.x, pp); pp = fmaf(q2.y, q2.y, pp); pp = fmaf(q2.z, q2.z, pp); pp = fmaf(q2.w, q2.w, pp);
      pp = fmaf(q3.x, q3.x, pp); pp = fmaf(q3.y, q3.y, pp); pp = fmaf(q3.z, q3.z, pp); pp = fmaf(q3.w, q3.w, pp);
      smem[L_PP + wv * 32 + ln] = pp;
    }
    #pragma unroll
    for (int g = 0; g < 2; ++g) {
      // ---- GEMM1: Dot tiles [16 samples x 64 attractors], K=16 ----
      v8f dt[4] = {{}, {}, {}, {}};
      #pragma unroll
      for (int c = 0; c < 4; ++c) {
        v2f a = *(const v2f*)&smem[pBase + (g * 16 + nIdx) * 16 + c * 4 + 2 * hi];
        #pragma unroll
        for (int t = 0; t < 4; ++t) dt[t] = WMMA4(a, B1[t][c], dt[t]);
      }
      // ---- elementwise: coef = G / (sqrt(d2)+1e-6)^2, scatter to A-layout LDS ----
      float ppv[8];
      #pragma unroll
      for (int i = 0; i < 8; ++i) ppv[i] = smem[L_PP + wv * 32 + g * 16 + i + 8 * hi];
      #pragma unroll
      for (int t = 0; t < 4; ++t) {
        #pragma unroll
        for (int i = 0; i < 8; ++i) {
          float d2 = fmaf(-2.f, dt[t][i], aaL[t] + ppv[i]);
          d2 = fmaxf(d2, 0.f);
          float dist = __builtin_amdgcn_sqrtf(d2) + 1e-6f;
          float coef = GRAV * __builtin_amdgcn_rcpf(dist * dist);
          smem[cBase + (g * 16 + i + 8 * hi) * 66 + t * 16 + nIdx] = coef;
        }
      }
      // ---- GEMM2: Sd[s,d], K=64; S0 row-sums ride along on the A-fragments ----
      v8f Ft = {};
      float s0p = 0.f;
      #pragma unroll
      for (int c = 0; c < 16; ++c) {
        v2f a = *(const v2f*)&smem[cBase + (g * 16 + nIdx) * 66 + c * 4 + 2 * hi];
        Ft = WMMA4(a, B2[c], Ft);
        s0p += a.x + a.y;
      }
      float s0sum = s0p + __shfl_xor(s0p, 16, 32);   // combine wave halves (K%4 {0,1} + {2,3})
      if (hi == 0) smem[L_S0 + wv * 32 + g * 16 + nIdx] = s0sum;
      // ---- update p += 0.1*(Sd - S0*p) in C-layout ----
      #pragma unroll
      for (int i = 0; i < 8; ++i) {
        int m = g * 16 + i + 8 * hi;
        float s0v  = smem[L_S0 + wv * 32 + m];
        float pold = smem[pBase + m * 16 + nIdx];
        float del  = 0.1f * fmaf(-s0v, pold, Ft[i]);
        smem[pBase + m * 16 + nIdx] = pold + del;
        if (it == 19) chg = fmaf(del, del, chg);
      }
    }
  }

  // ================= deterministic block partial sums =================
  smem[L_CHG + tid] = chg;
  __syncthreads();
  if (tid < 16) {
    float s = 0.f;
    for (int r = 0; r < 128; ++r) s += smem[L_P + r * 16 + tid];
    af_part[bk * 16 + tid] = s;
  } else if (tid == 16) {
    float s = 0.f;
    for (int r = 0; r < 128; ++r) s += smem[L_CHG + r];
    chg_part[bk] = s;
  }
}

__global__ __launch_bounds__(256) void gda_tail(
    const float* __restrict__ af_part, const float* __restrict__ chg_part,
    const float* __restrict__ dw1, const float* __restrict__ db1,
    const float* __restrict__ dw2, const float* __restrict__ db2,
    const float* __restrict__ mw,  const float* __restrict__ mb,
    float* __restrict__ out, float* __restrict__ rec_ws)
{
  __shared__ float afs[512];
  __shared__ float hds[512];
  const int tid = threadIdx.x;

  for (int idx = tid; idx < 512; idx += 256) {
    int b = idx >> 4, d = idx & 15;
    float s = 0.f;
    for (int k = 0; k < 16; ++k) s += af_part[((b << 4) + k) * 16 + d];
    float af = s * (1.0f / 2048.0f);
    afs[idx] = af;
    out[AF_OFF + idx] = af;
  }
  if (tid == 0) {
    float s = 0.f;
    for (int i = 0; i < NBLK; ++i) s += chg_part[i];
    out[C_OFF] = sqrtf(s);
  }
  __syncthreads();

  for (int idx = tid; idx < 512; idx += 256) {
    int b = idx >> 4, h = idx & 15;
    float acc = db1[h];
    #pragma unroll
    for (int d = 0; d < 16; ++d) acc = fmaf(afs[(b << 4) + d], dw1[d * 16 + h], acc);
    hds[idx] = fmaxf(acc, 0.f);
  }
  if (tid < 32) {
    float acc = mb[0];
    #pragma unroll
    for (int d = 0; d < 16; ++d) acc = fmaf(afs[(tid << 4) + d], mw[d], acc);
    out[M_OFF + tid] = 1.0f / (1.0f + expf(-acc));
  }
  __syncthreads();

  for (int idx = tid; idx < B_ * IN_; idx += 256) {
    int b = idx / IN_, o = idx - b * IN_;
    float acc = db2[o];
    #pragma unroll
    for (int h = 0; h < 16; ++h) acc = fmaf(hds[(b << 4) + h], dw2[h * IN_ + o], acc);
    rec_ws[idx] = acc;
  }
}

// broadcast rec[32,300] -> out[32,2048,300]; float4 stores, rows are 75 float4s
__global__ __launch_bounds__(256) void gda_bcast(const float* __restrict__ rec_ws,
                                                 float* __restrict__ out)
{
  unsigned idx4 = blockIdx.x * 256u + threadIdx.x;   // 0 .. 4,915,199
  unsigned row  = idx4 / 75u;                        // flat (b*2048 + s)
  unsigned o4   = (idx4 - row * 75u) * 4u;
  unsigned b    = row >> 11;                         // /2048
  float4 v = *(const float4*)(rec_ws + b * 300u + o4);
  *(float4*)(out + REC_OFF + (size_t)idx4 * 4u) = v;
}

extern "C" void kernel_launch(void* const* d_in, const int* in_sizes, int n_in,
                              void* d_out, int out_size, void* d_ws, size_t ws_size,
                              hipStream_t stream) {
  const float* X   = (const float*)d_in[0];
  const float* W1  = (const float*)d_in[1];
  const float* b1  = (const float*)d_in[2];
  const float* W2  = (const float*)d_in[3];
  const float* b2  = (const float*)d_in[4];
  const float* att = (const float*)d_in[5];
  const float* dw1 = (const float*)d_in[6];
  const float* db1 = (const float*)d_in[7];
  const float* dw2 = (const float*)d_in[8];
  const float* db2 = (const float*)d_in[9];
  const float* mw  = (const float*)d_in[10];
  const float* mb  = (const float*)d_in[11];
  float* out = (float*)d_out;
  float* ws  = (float*)d_ws;

  float* af_part  = ws + WS_AFPART;
  float* chg_part = ws + WS_CHGPART;
  float* rec_ws   = ws + WS_REC;

  gda_main<<<NBLK, 128, 0, stream>>>(X, W1, b1, W2, b2, att, af_part, chg_part);
  gda_tail<<<1, 256, 0, stream>>>(af_part, chg_part, dw1, db1, dw2, db2, mw, mb, out, rec_ws);
  gda_bcast<<<19200, 256, 0, stream>>>(rec_ws, out);   // 4,915,200 float4 stores
}